// SIP_21938692948270
// MI455X (gfx1250) — compile-verified
//
#include <hip/hip_runtime.h>
#include <hip/hip_bf16.h>

// ---------------------------------------------------------------------------
// SSN iteration (superpixel affinity) for MI455X / gfx1250.
// B=4, C=32 (+32 contrast), H=W=256, 16x16 cells -> nsp=256, n=65536, Ct=64.
// Distance core runs on V_WMMA_F32_16X16X4_F32 (fp32 matrix engine).
// HBM-bound by the 256 MB abs_aff zero-fill -> b128 stores for the fill.
// ---------------------------------------------------------------------------

typedef __attribute__((ext_vector_type(2))) float v2f;
typedef __attribute__((ext_vector_type(8))) float v8f;

#define BB       4
#define CCH      32          // channels per tensor (x, x_contrast)
#define CT       64          // total channels
#define HW       256         // H == W
#define NSPW     16          // cells per row/col
#define NSP      256         // cells per image
#define NPIX     65536       // pixels per image
#define CELLPIX  256         // pixels per cell
#define NTOT     (4LL * 256LL * 65536LL)   // abs_aff element count

// ---------------------------------------------------------------------------
// Vectorized zero-fill: b128 stores (512B per wave per instruction).
// ---------------------------------------------------------------------------
__global__ void ssn_zero4(float* __restrict__ out, long long count) {
    long long tid    = (long long)blockIdx.x * blockDim.x + threadIdx.x;
    long long stride = (long long)gridDim.x * blockDim.x;
    long long n4     = count >> 2;
    float4 z = {0.0f, 0.0f, 0.0f, 0.0f};
    float4* __restrict__ out4 = (float4*)out;
    for (long long i = tid; i < n4; i += stride) out4[i] = z;
    for (long long i = (n4 << 2) + tid; i < count; i += stride) out[i] = 0.0f;
}

// Plant num_spixels (=256) scalar appended after abs_aff.
__global__ void ssn_scalar(float* __restrict__ out) {
    if (threadIdx.x == 0 && blockIdx.x == 0) out[NTOT] = 256.0f;
}

// ---------------------------------------------------------------------------
// Cell means: spix[b, c, cell] = mean over 16x16 patch. One block per (b,cell),
// 8 waves; wave w reduces channels w*8..w*8+7 with wave-level shuffles only.
// ---------------------------------------------------------------------------
__global__ __launch_bounds__(256) void ssn_means(const float* __restrict__ x,
                                                 const float* __restrict__ xc,
                                                 float* __restrict__ spix) {
    int blk  = blockIdx.x;
    int b    = blk >> 8;
    int cell = blk & 255;
    int sy = cell >> 4, sx = cell & 15;

    int t = threadIdx.x, lane = t & 31, wave = t >> 5;

    for (int ci = 0; ci < 8; ++ci) {
        int c = wave * 8 + ci;
        const float* src = (c < CCH) ? x : xc;
        int cc = c & 31;
        const float* base = src + ((size_t)(b * CCH + cc)) * NPIX
                                + (size_t)(sy * 16) * HW + sx * 16;
        float s = 0.0f;
#pragma unroll
        for (int ch = 0; ch < 8; ++ch) {
            int p = ch * 32 + lane;                 // pixel in cell
            s += base[(p >> 4) * HW + (p & 15)];    // coalesced: 2x64B per wave
        }
#pragma unroll
        for (int off = 16; off > 0; off >>= 1) s += __shfl_down(s, off, 32);
        if (lane == 0) spix[(b * CT + c) * NSP + cell] = s * (1.0f / 256.0f);
    }
}

// ---------------------------------------------------------------------------
// One SSN iteration for one (b, cell). Distances via fp32 WMMA:
//   dist_j = ||p||^2_w - 2 * p . (w * s_j) + ||s_j||^2_w
// final_iter==0: accumulate num/den (centroid update) with global atomics.
// final_iter==1: scatter softmax affinities into abs_aff (d_out).
// ---------------------------------------------------------------------------
__global__ __launch_bounds__(256) void ssn_iter(const float* __restrict__ x,
                                                const float* __restrict__ xc,
                                                const float* __restrict__ spix,
                                                float* __restrict__ num_ws,
                                                float* __restrict__ out,
                                                int final_iter) {
    int blk  = blockIdx.x;
    int b    = blk >> 8;
    int cell = blk & 255;
    int sy = cell >> 4, sx = cell & 15;

    __shared__ float cand_w[CT][16];      // weighted candidate features (w*s)
    __shared__ float sqn[16];             // ||s_j||^2_w
    __shared__ float dist_lds[CELLPIX][17];
    __shared__ float aff_lds[9][CELLPIX];

    int t = threadIdx.x, lane = t & 31, wave = t >> 5;
    int half = lane >> 4;                 // 0: K rows {0,1}; 1: K rows {2,3}
    int lm   = lane & 15;                 // pixel-in-tile (A) / candidate col (B)

    // Candidate list (uniform across the block; cheap to recompute per thread).
    int  cand_id[9];
    bool validj[9];
#pragma unroll
    for (int j = 0; j < 9; ++j) {
        int dy = j / 3 - 1, dx = j % 3 - 1;
        int ny = sy + dy, nx = sx + dx;
        validj[j] = (nx >= 0) & (nx < NSPW) & (ny >= 0) & (ny < NSPW);
        int flat = ny * NSPW + nx;
        cand_id[j] = flat < 0 ? 0 : (flat > NSP - 1 ? NSP - 1 : flat);
    }

    // ---- Phase 0: stage weighted candidate features + their weighted norms.
    for (int idx = t; idx < CT * 16; idx += 256) {
        int c = idx >> 4, j = idx & 15;
        float v = 0.0f;
        if (j < 9) {
            float w = (c < CCH) ? 1.0f : 10.0f;
            v = w * spix[(b * CT + c) * NSP + cand_id[j]];
        }
        cand_w[c][j] = v;
    }
    __syncthreads();
    if (t < 16) {
        float s = 0.0f;
        for (int c = 0; c < CT; ++c) {
            float invw = (c < CCH) ? 1.0f : 0.1f;
            float v = cand_w[c][t];       // = w*s  -> w*s^2 = v*v/w
            s += v * v * invw;
        }
        sqn[t] = s;
    }
    __syncthreads();

    // ---- Phase 1: per-wave WMMA GEMM. Wave handles 2 M-tiles (patch rows).
    for (int i = 0; i < 2; ++i) {
        int mt  = wave * 2 + i;                     // patch row 0..15
        int gy  = sy * 16 + mt;
        int gx0 = sx * 16;
        v8f  acc  = {};
        float pw2 = 0.0f;
#pragma unroll
        for (int kk = 0; kk < 16; ++kk) {           // K = 64 in steps of 4
            int c0 = kk * 4 + half * 2;             // channel of A.x for this lane
            const float* src = (c0 < CCH) ? x : xc;
            int cc = c0 & 31;
            size_t ga = ((size_t)(b * CCH + cc)) * NPIX + (size_t)gy * HW + gx0 + lm;
            v2f a;
            a.x = src[ga];                          // channel c0   (64B/half-wave)
            a.y = src[ga + NPIX];                   // channel c0+1
            float w = (kk < 8) ? 1.0f : 10.0f;      // kk>=8 -> contrast channels
            pw2 += w * (a.x * a.x + a.y * a.y);
            v2f bf;
            bf.x = cand_w[kk * 4 + half * 2 + 0][lm];
            bf.y = cand_w[kk * 4 + half * 2 + 1][lm];
            acc = __builtin_amdgcn_wmma_f32_16x16x4_f32(
                      false, a, false, bf, (short)0, acc, false, false);
        }
        // combine per-half channel partials -> full ||p||^2_w (held at lane m)
        float pw2full = pw2 + __shfl_xor(pw2, 16, 32);
        // C layout: VGPR r, lanes 0-15 -> M=r; lanes 16-31 -> M=r+8; N = lane&15
#pragma unroll
        for (int r = 0; r < 8; ++r) {
            int M = r + half * 8;
            float pm = __shfl(pw2full, M, 32);
            dist_lds[mt * 16 + M][lm] = pm - 2.0f * acc[r] + sqn[lm];
        }
    }
    __syncthreads();

    // ---- Phase 2: per-pixel softmax over the 9 (uniform-valid) candidates.
    {
        int p = t;
        float d[9];
#pragma unroll
        for (int j = 0; j < 9; ++j) d[j] = dist_lds[p][j];
        float mn = 3.0e38f;
#pragma unroll
        for (int j = 0; j < 9; ++j) if (validj[j]) mn = fminf(mn, d[j]);
        float e[9], ssum = 0.0f;
#pragma unroll
        for (int j = 0; j < 9; ++j) {
            e[j] = validj[j] ? __expf(mn - d[j]) : 0.0f;
            ssum += e[j];
        }
        float inv = 1.0f / ssum;                    // center candidate always valid

        if (final_iter) {
            int gy = sy * 16 + (p >> 4), gx = sx * 16 + (p & 15);
            size_t col = (size_t)gy * HW + gx;
#pragma unroll
            for (int j = 0; j < 9; ++j)             // coalesced 64B row chunks
                if (validj[j])
                    out[((size_t)b * NSP + cand_id[j]) * (size_t)NPIX + col] = e[j] * inv;
        } else {
#pragma unroll
            for (int j = 0; j < 9; ++j) aff_lds[j][p] = e[j] * inv;
        }
    }

    // ---- Phase 3 (iter 0 only): centroid update partial sums -> atomics.
    if (!final_iter) {
        __syncthreads();
        for (int pair = t; pair < 9 * 65; pair += 256) {
            int j = pair / 65, c = pair - j * 65;   // c==64 -> denominator
            if (!validj[j]) continue;
            float s = 0.0f;
            if (c < CT) {
                const float* src = (c < CCH) ? x : xc;
                int cc = c & 31;
                const float* base = src + ((size_t)(b * CCH + cc)) * NPIX
                                        + (size_t)(sy * 16) * HW + sx * 16;
                for (int p = 0; p < CELLPIX; ++p)
                    s += aff_lds[j][p] * base[(p >> 4) * HW + (p & 15)];
            } else {
                for (int p = 0; p < CELLPIX; ++p) s += aff_lds[j][p];
            }
            atomicAdd(&num_ws[((size_t)b * NSP + cand_id[j]) * 65 + c], s);
        }
    }
}

// ---------------------------------------------------------------------------
// spix1[b, c, s] = num[b, s, c] / (den[b, s] + 1e-16)
// ---------------------------------------------------------------------------
__global__ __launch_bounds__(256) void ssn_finalize(const float* __restrict__ num_ws,
                                                    float* __restrict__ spix1) {
    int i = blockIdx.x * blockDim.x + threadIdx.x;   // over B*CT*NSP = 65536
    if (i >= BB * CT * NSP) return;
    int s = i & 255;
    int c = (i >> 8) & 63;
    int b = i >> 14;
    const float* row = num_ws + ((size_t)b * NSP + s) * 65;
    spix1[(b * CT + c) * NSP + s] = row[c] / (row[64] + 1e-16f);
}

// ---------------------------------------------------------------------------
extern "C" void kernel_launch(void* const* d_in, const int* in_sizes, int n_in,
                              void* d_out, int out_size, void* d_ws, size_t ws_size,
                              hipStream_t stream) {
    const float* x  = (const float*)d_in[0];
    const float* xc = (const float*)d_in[1];
    float* out = (float*)d_out;

    // Workspace layout (floats): spix0 | num/den | spix1
    float* spix0  = (float*)d_ws;                       // 4*64*256      = 65536
    float* num_ws = spix0 + BB * CT * NSP;              // 4*256*65      = 66560
    float* spix1  = num_ws + BB * NSP * 65;             // 4*64*256      = 65536
    (void)ws_size; (void)in_sizes; (void)n_in;

    // 1) Zero abs_aff output (b128 stores; dominant HBM phase).
    ssn_zero4<<<8192, 256, 0, stream>>>(out, (long long)out_size);
    if ((long long)out_size > NTOT)      // plant num_spixels = 256 tail scalar
        ssn_scalar<<<1, 64, 0, stream>>>(out);
    // 2) Zero centroid-update accumulators.
    ssn_zero4<<<64, 256, 0, stream>>>(num_ws, (long long)(BB * NSP * 65));
    // 3) Initial centroids (cell means).
    ssn_means<<<BB * NSP, 256, 0, stream>>>(x, xc, spix0);
    // 4) Iteration 0: WMMA distances -> softmax -> num/den accumulation.
    ssn_iter<<<BB * NSP, 256, 0, stream>>>(x, xc, spix0, num_ws, out, 0);
    // 5) Centroid update.
    ssn_finalize<<<(BB * CT * NSP + 255) / 256, 256, 0, stream>>>(num_ws, spix1);
    // 6) Iteration 1: WMMA distances -> softmax -> scatter abs_aff.
    ssn_iter<<<BB * NSP, 256, 0, stream>>>(x, xc, spix1, num_ws, out, 1);
}